// PeronaMalikLayer_15427522527845
// MI455X (gfx1250) — compile-verified
//
#include <hip/hip_runtime.h>
#include <stdint.h>

// Problem dims (fixed by the reference)
#define BATCH 32
#define IMH 512
#define IMW 512
#define HWPX (IMH * IMW)            // 262144 pixels per image
#define NPIX ((size_t)BATCH * HWPX) // 8,388,608 luminance elements

// Tiling for the diffusion step
#define TILE 64
#define LW (TILE + 2)   // 66: tile + 1px halo each side
#define LSTRIDE LW      // LDS pitch == 66 so TDM contiguous rows match layout
#define NTHREADS 256    // 8 wave32 per block
#define ROWS_PER_THREAD 16

typedef unsigned int u32x4 __attribute__((ext_vector_type(4)));
typedef int i32x4 __attribute__((ext_vector_type(4)));
typedef int i32x8 __attribute__((ext_vector_type(8)));

// Toolchain arity dispatch for the TDM builtin (bridge doc):
// therock headers present -> 6-arg form; plain ROCm 7.2 -> 5-arg form.
#if __has_include(<hip/amd_detail/amd_gfx1250_TDM.h>)
#define TDM_LOAD_TO_LDS(g0, g1, g2, g3)                                        \
  do {                                                                         \
    i32x8 _z8 = {0, 0, 0, 0, 0, 0, 0, 0};                                      \
    __builtin_amdgcn_tensor_load_to_lds((g0), (g1), (g2), (g3), _z8, 0);       \
  } while (0)
#else
#define TDM_LOAD_TO_LDS(g0, g1, g2, g3)                                        \
  __builtin_amdgcn_tensor_load_to_lds((g0), (g1), (g2), (g3), 0)
#endif

// Low 32 bits of a flat LDS pointer == LDS byte offset (aperture in high bits).
__device__ __forceinline__ unsigned lds_byte_off(const void* p) {
  return (unsigned)(uintptr_t)p;
}

// ---------------------------------------------------------------------------
// One Perona-Malik diffusion step (8-direction stencil, zero 'SAME' pad).
// Tile staged into LDS via the CDNA5 Tensor Data Mover when the halo origin
// is in-bounds (49/64 tiles); per-lane async global->LDS loads otherwise.
// Placed first in the file so the disasm snippet shows the TDM/async ops.
// ---------------------------------------------------------------------------
__global__ __launch_bounds__(NTHREADS) void pm_step(const float* __restrict__ src,
                                                    float* __restrict__ dst,
                                                    const float* __restrict__ pdelta,
                                                    const float* __restrict__ pkappa) {
  __shared__ float tile[LW * LSTRIDE];

  const int bx = blockIdx.x;       // 0..7  tile col
  const int by = blockIdx.y;       // 0..7  tile row
  const int b  = blockIdx.z;       // batch
  const int gx0 = bx * TILE;
  const int gy0 = by * TILE;
  const float* srcb = src + (size_t)b * HWPX;
  const int tid = threadIdx.x;

  const bool tdm_path = (bx >= 1) && (by >= 1); // top/left halo in-bounds

  if (tdm_path) {
    // One TDM 2D-tile DMA per block, issued by wave 0 only. Right/bottom
    // zero padding comes from TDM OOB zero-fill: tensor_dim0/1 are the
    // remaining extent measured from the tile origin (ISA 8.2).
    if (tid < 32) {
      uint64_t ga =
          (uint64_t)(uintptr_t)(srcb + (size_t)(gy0 - 1) * IMW + (gx0 - 1));
      unsigned ldsa = lds_byte_off(&tile[0]);
      unsigned rem0 = (unsigned)(IMW - (gx0 - 1)); // cols to tensor edge
      unsigned rem1 = (unsigned)(IMH - (gy0 - 1)); // rows to tensor edge

      u32x4 g0;
      g0[0] = 1u;                                   // count=1 (valid user D#)
      g0[1] = ldsa;                                 // lds_addr (bytes)
      g0[2] = (unsigned)(ga & 0xffffffffu);         // global_addr[31:0]
      g0[3] = (unsigned)((ga >> 32) & 0x01ffffffu)  // global_addr[56:32]
              | (2u << 30);                         // type=2 ("image")

      i32x8 g1;
      g1[0] = (int)(2u << 16);                      // data_size=4B; no pad/iter
      g1[1] = (int)((rem0 & 0xffffu) << 16);        // tensor_dim0[15:0]
      g1[2] = (int)(((rem0 >> 16) & 0xffffu) | ((rem1 & 0xffffu) << 16));
      g1[3] = (int)(((rem1 >> 16) & 0xffffu) | ((unsigned)LW << 16)); // tile_dim0
      g1[4] = LW;                                   // tile_dim1=66, tile_dim2=0
      g1[5] = IMW;                                  // tensor_dim0_stride=512
      g1[6] = 0;
      g1[7] = 0;

      i32x4 g2 = {1, 1, 0, 0};          // tensor_dim2=1, tensor_dim3=1 (benign)
      i32x4 g3 = {0, 0x00010000, 0, 0}; // tensor_dim4=1 (benign)

      TDM_LOAD_TO_LDS(g0, g1, g2, g3);
      __builtin_amdgcn_s_wait_tensorcnt(0);
    }
  } else {
    // Boundary tiles (bx==0 || by==0): per-lane async loads for in-bounds
    // cells, zero DS stores for OOB halo cells (disjoint LDS addresses, so
    // the unordered ASYNCcnt writes and DScnt stores cannot race).
    for (int c = tid; c < LW * LW; c += NTHREADS) {
      int ly = c / LW;
      int lx = c - ly * LW;
      int gy = gy0 + ly - 1;
      int gx = gx0 + lx - 1;
      float* lp = &tile[ly * LSTRIDE + lx];
      if ((unsigned)gy < (unsigned)IMH && (unsigned)gx < (unsigned)IMW) {
        unsigned loff = lds_byte_off(lp);
        const float* gp = srcb + (size_t)gy * IMW + gx;
        asm volatile("global_load_async_to_lds_b32 %0, %1, off"
                     :
                     : "v"(loff), "v"(gp)
                     : "memory");
      } else {
        *lp = 0.0f; // zero 'SAME' padding
      }
    }
    asm volatile("s_wait_asynccnt 0x0" ::: "memory");
  }
  __syncthreads();

  const float delta = *pdelta;
  const float kappa = *pkappa;
  const float ik2 = __builtin_amdgcn_rcpf(kappa * kappa);

  // Each thread sweeps a contiguous 16-row strip at one column, rotating the
  // 3x3 neighborhood through registers: 3 LDS reads per pixel instead of 9.
  const int tx  = tid & (TILE - 1);            // 0..63 column
  const int ry0 = (tid >> 6) * ROWS_PER_THREAD; // 0,16,32,48
  float* dstb = dst + (size_t)b * HWPX;

  // g(n) = n / (1 + n^2/kappa^2), via fast v_rcp_f32
  auto g = [ik2](float n) {
    float d = __builtin_fmaf(n * n, ik2, 1.0f);
    return n * __builtin_amdgcn_rcpf(d);
  };

  int base = ry0 * LSTRIDE + (tx + 1); // tile row ry0 = image row gy0+ry0-1
  float a0 = tile[base - 1], a1 = tile[base], a2 = tile[base + 1];
  base += LSTRIDE;
  float b0 = tile[base - 1], b1 = tile[base], b2 = tile[base + 1];

#pragma unroll
  for (int k = 0; k < ROWS_PER_THREAD; ++k) {
    base += LSTRIDE;
    float c0 = tile[base - 1], c1 = tile[base], c2 = tile[base + 1];

    float cc = b1;
    float axial = g(a1 - cc) + g(c1 - cc) + g(b2 - cc) + g(b0 - cc);
    float diag  = g(a0 - cc) + g(a2 - cc) + g(c0 - cc) + g(c2 - cc);
    float upd   = __builtin_fmaf(0.5f, diag, axial);

    dstb[(size_t)(gy0 + ry0 + k) * IMW + (gx0 + tx)] =
        __builtin_fmaf(delta, upd, cc);

    a0 = b0; a1 = b1; a2 = b2;
    b0 = c0; b1 = c1; b2 = c2;
  }
}

// ---------------------------------------------------------------------------
// RGB -> luminance
// ---------------------------------------------------------------------------
__global__ __launch_bounds__(NTHREADS) void pm_lum(const float* __restrict__ x,
                                                   float* __restrict__ u) {
  size_t i = (size_t)blockIdx.x * NTHREADS + threadIdx.x;
  if (i >= NPIX) return;
  size_t b = i / HWPX;
  size_t p = i - b * HWPX;
  const float* xb = x + b * (size_t)(3 * HWPX);
  float r = xb[p];
  float g = xb[HWPX + p];
  float bl = xb[2 * (size_t)HWPX + p];
  u[i] = __builtin_fmaf(0.299f, r, __builtin_fmaf(0.587f, g, 0.114f * bl));
}

// ---------------------------------------------------------------------------
// out[b,o,h,w] = (sum_i W[o,i]) * u[b,h,w] + bias[o]
// (u repeated to 3 identical channels makes the 1x1 conv a scalar scale)
// ---------------------------------------------------------------------------
__global__ __launch_bounds__(NTHREADS) void pm_expand(const float* __restrict__ u,
                                                      const float* __restrict__ Wexp,
                                                      const float* __restrict__ bexp,
                                                      float* __restrict__ out) {
  size_t i = (size_t)blockIdx.x * NTHREADS + threadIdx.x;
  if (i >= NPIX) return;
  size_t b = i / HWPX;
  size_t p = i - b * HWPX;
  float uv = u[i];
  float* ob = out + b * (size_t)(3 * HWPX);
#pragma unroll
  for (int o = 0; o < 3; ++o) {
    float s = Wexp[o * 3 + 0] + Wexp[o * 3 + 1] + Wexp[o * 3 + 2];
    ob[(size_t)o * HWPX + p] = __builtin_fmaf(s, uv, bexp[o]);
  }
}

// ---------------------------------------------------------------------------
extern "C" void kernel_launch(void* const* d_in, const int* in_sizes, int n_in,
                              void* d_out, int out_size, void* d_ws, size_t ws_size,
                              hipStream_t stream) {
  const float* x      = (const float*)d_in[0];
  const float* pdelta = (const float*)d_in[1];
  const float* pkappa = (const float*)d_in[2];
  const float* Wexp   = (const float*)d_in[3];
  const float* bexp   = (const float*)d_in[4];
  float* out = (float*)d_out;

  const size_t uBytes = NPIX * sizeof(float);
  float* bufA = (float*)d_ws; // final u lands here after an even # of steps
  float* bufB = (ws_size >= 2 * uBytes) ? (bufA + NPIX) : (float*)d_out;

  const int pixBlocks = (int)((NPIX + NTHREADS - 1) / NTHREADS);

  pm_lum<<<pixBlocks, NTHREADS, 0, stream>>>(x, bufA);

  dim3 grid(IMW / TILE, IMH / TILE, BATCH); // 8 x 8 x 32 = 2048 blocks
  float* s = bufA;
  float* d = bufB;
  for (int it = 0; it < 30; ++it) {
    pm_step<<<grid, NTHREADS, 0, stream>>>(s, d, pdelta, pkappa);
    float* t = s; s = d; d = t;
  }
  // 30 steps (even) -> result is back in bufA (== s), which lives in d_ws,
  // so pm_expand never reads memory it is concurrently overwriting.
  pm_expand<<<pixBlocks, NTHREADS, 0, stream>>>(s, Wexp, bexp, out);
}